// BoundaryConvLayer_61400852463787
// MI455X (gfx1250) — compile-verified
//
#include <hip/hip_runtime.h>
#include <hip/hip_bf16.h>

typedef float v2f __attribute__((ext_vector_type(2)));
typedef float v8f __attribute__((ext_vector_type(8)));

#define SX 132   // LDS row stride for x tile (conflict-free, 8B-aligned A-frag reads)
#define SH 260   // LDS row stride for h tile
#define ROWS 32  // node rows per block = two 16-row WMMA tiles (B-fragment reuse x2)

__device__ __forceinline__ v8f wmma_f32(v2f a, v2f b, v8f c) {
  // D = A(16x4,f32) * B(4x16,f32) + C(16x16,f32)
  return __builtin_amdgcn_wmma_f32_16x16x4_f32(
      /*neg_a=*/false, a, /*neg_b=*/false, b,
      /*c_mod=*/(short)0, c, /*reuse_a=*/false, /*reuse_b=*/false);
}

__device__ __forceinline__ float softplus_f(float v) {
  return v > 20.0f ? v : log1pf(expf(v));
}

__device__ __forceinline__ float wave_sum(float v) {
#pragma unroll
  for (int off = 16; off > 0; off >>= 1) v += __shfl_xor(v, off, 32);
  return v;
}

// ---------------------------------------------------------------------------
// Fused GEMM kernel: per 32-row node tile (2 x 16-row WMMA tiles) computes
//   z     = x @ W_fc^T + b_fc                  -> zbuf   [N,128]
//   rate  = softplus(x @ W_rate^T)             -> ratebuf[N,128]
//   h     = softplus(x @ W1^T + b1)            -> LDS    [32,256]
//   gpre  = h @ W2^T + b2                      -> gpre   [N,128]
// 256 threads = 8 waves; wave w owns 64 of the 512 concatenated phase-1 cols.
// Each B fragment (weights) feeds TWO wmmas (one per row tile).
// ---------------------------------------------------------------------------
__global__ __launch_bounds__(256)
void gemm_fused_kernel(const float* __restrict__ x,
                       const float* __restrict__ W_fc, const float* __restrict__ b_fc,
                       const float* __restrict__ W_rate,
                       const float* __restrict__ W1, const float* __restrict__ b1,
                       const float* __restrict__ W2, const float* __restrict__ b2,
                       float* __restrict__ zbuf, float* __restrict__ ratebuf,
                       float* __restrict__ gpre, int Nn) {
  __shared__ float lx[ROWS * SX];
  __shared__ float lh[ROWS * SH];
  const int tid  = threadIdx.x;
  const int wave = tid >> 5;
  const int lane = tid & 31;
  const int row0 = blockIdx.x * ROWS;

  // Stage x tile (coalesced), zero-pad tail rows
  for (int i = tid; i < ROWS * 128; i += 256) {
    int r = i >> 7, c = i & 127;
    float v = 0.0f;
    if (row0 + r < Nn) v = x[(size_t)(row0 + r) * 128 + c];
    lx[r * SX + c] = v;
  }
  __syncthreads();

  const int m    = lane & 15;        // M (A) / N (B,C,D) position within tile
  const int kh   = (lane >> 4) << 1; // K sub-offset: 0 for lanes 0-15, 2 for 16-31
  const int mrow = lane >> 4;        // C/D row offset selector (0 -> j, 1 -> j+8)

  // ---- phase 1: [z(128) | rate(128) | h(256)], K = 128 ----
  {
    const int wcb = wave * 64;
    const float* Wp;
    int ob, mode;
    if (wcb < 128)       { Wp = W_fc;   ob = wcb;       mode = 0; }
    else if (wcb < 256)  { Wp = W_rate; ob = wcb - 128; mode = 1; }
    else                 { Wp = W1;     ob = wcb - 256; mode = 2; }

    v8f acc[2][4] = {};
#pragma unroll 2
    for (int kk = 0; kk < 128; kk += 4) {
      v2f a0 = *(const v2f*)&lx[m * SX + kk + kh];
      v2f a1 = *(const v2f*)&lx[(16 + m) * SX + kk + kh];
      v2f bfr[4];
#pragma unroll
      for (int t = 0; t < 4; ++t)
        bfr[t] = *(const v2f*)&Wp[(size_t)(ob + t * 16 + m) * 128 + kk + kh];
#pragma unroll
      for (int t = 0; t < 4; ++t) {
        acc[0][t] = wmma_f32(a0, bfr[t], acc[0][t]);
        acc[1][t] = wmma_f32(a1, bfr[t], acc[1][t]);
      }
    }

#pragma unroll
    for (int rt = 0; rt < 2; ++rt) {
#pragma unroll
      for (int t = 0; t < 4; ++t) {
        const int col = ob + t * 16 + m;
#pragma unroll
        for (int j = 0; j < 8; ++j) {
          const int r = rt * 16 + j + 8 * mrow;
          float val = acc[rt][t][j];
          if (mode == 0) {
            val += b_fc[col];
            if (row0 + r < Nn) zbuf[(size_t)(row0 + r) * 128 + col] = val;
          } else if (mode == 1) {
            val = softplus_f(val);
            if (row0 + r < Nn) ratebuf[(size_t)(row0 + r) * 128 + col] = val;
          } else {
            lh[r * SH + col] = softplus_f(val + b1[col]);
          }
        }
      }
    }
  }
  __syncthreads();

  // ---- phase 2: gamma_pre = h @ W2^T + b2, K = 256, 16 cols per wave ----
  {
    const int cb = wave * 16;
    v8f acc[2] = {};
#pragma unroll 2
    for (int kk = 0; kk < 256; kk += 4) {
      v2f a0 = *(const v2f*)&lh[m * SH + kk + kh];
      v2f a1 = *(const v2f*)&lh[(16 + m) * SH + kk + kh];
      v2f b  = *(const v2f*)&W2[(size_t)(cb + m) * 256 + kk + kh];
      acc[0] = wmma_f32(a0, b, acc[0]);
      acc[1] = wmma_f32(a1, b, acc[1]);
    }
    const float bb = b2[cb + m];
#pragma unroll
    for (int rt = 0; rt < 2; ++rt) {
#pragma unroll
      for (int j = 0; j < 8; ++j) {
        const int r = rt * 16 + j + 8 * mrow;
        if (row0 + r < Nn) gpre[(size_t)(row0 + r) * 128 + cb + m] = acc[rt][j] + bb;
      }
    }
  }
}

// ---------------------------------------------------------------------------
// Edge scatter: one wave per edge. agg[row] += z[col]; cnt[row] += 1.
// (cnt[row]*z[row] is folded in later -> halves gather traffic.)
// ---------------------------------------------------------------------------
__global__ __launch_bounds__(256)
void edge_kernel(const long long* __restrict__ rows,
                 const long long* __restrict__ cols,
                 const float* __restrict__ zbuf,
                 float* __restrict__ agg, int* __restrict__ cnt, int E) {
  const int gw   = (int)((blockIdx.x * (size_t)blockDim.x + threadIdx.x) >> 5);
  const int lane = threadIdx.x & 31;
  if (gw >= E) return;
  const long long r = rows[gw];
  const long long c = cols[gw];
  float4 v = ((const float4*)(zbuf + (size_t)c * 128))[lane];
  float* ar = agg + (size_t)r * 128 + lane * 4;
  atomicAdd(ar + 0, v.x);
  atomicAdd(ar + 1, v.y);
  atomicAdd(ar + 2, v.z);
  atomicAdd(ar + 3, v.w);
  if (lane == 0) atomicAdd(cnt + r, 1);
}

// ---------------------------------------------------------------------------
// Final per-node kernel: one wave per node (4 cols per lane).
// gamma = LN1(gpre); out = LN2((rate*(cnt*z+agg)+gamma)/(1+rate*deg+EPS) - z)
// ---------------------------------------------------------------------------
__global__ __launch_bounds__(256)
void final_kernel(const float* __restrict__ zbuf, const float* __restrict__ ratebuf,
                  const float* __restrict__ gpre, const float* __restrict__ agg,
                  const int* __restrict__ cnt, const float* __restrict__ degree,
                  const float* __restrict__ ln1g, const float* __restrict__ ln1b,
                  const float* __restrict__ ln2g, const float* __restrict__ ln2b,
                  float* __restrict__ out, int Nn) {
  const int gw   = (int)((blockIdx.x * (size_t)blockDim.x + threadIdx.x) >> 5);
  const int lane = threadIdx.x & 31;
  if (gw >= Nn) return;
  const size_t base = (size_t)gw * 128 + lane * 4;

  // LN1 over gamma_pre
  float4 g = *(const float4*)(gpre + base);
  float mean = wave_sum(g.x + g.y + g.z + g.w) * (1.0f / 128.0f);
  float d0 = g.x - mean, d1 = g.y - mean, d2 = g.z - mean, d3 = g.w - mean;
  float var = wave_sum(d0 * d0 + d1 * d1 + d2 * d2 + d3 * d3) * (1.0f / 128.0f);
  float inv = rsqrtf(var + 1e-5f);
  float4 w1v = *(const float4*)(ln1g + lane * 4);
  float4 b1v = *(const float4*)(ln1b + lane * 4);
  float gm0 = d0 * inv * w1v.x + b1v.x;
  float gm1 = d1 * inv * w1v.y + b1v.y;
  float gm2 = d2 * inv * w1v.z + b1v.z;
  float gm3 = d3 * inv * w1v.w + b1v.w;

  float4 zz = *(const float4*)(zbuf + base);
  float4 rr = *(const float4*)(ratebuf + base);
  float4 aa = *(const float4*)(agg + base);
  const float cf = (float)cnt[gw];
  const float dg = degree[gw];

  float o0 = (rr.x * (cf * zz.x + aa.x) + gm0) / (1.0f + rr.x * dg + 1e-4f) - zz.x;
  float o1 = (rr.y * (cf * zz.y + aa.y) + gm1) / (1.0f + rr.y * dg + 1e-4f) - zz.y;
  float o2 = (rr.z * (cf * zz.z + aa.z) + gm2) / (1.0f + rr.z * dg + 1e-4f) - zz.z;
  float o3 = (rr.w * (cf * zz.w + aa.w) + gm3) / (1.0f + rr.w * dg + 1e-4f) - zz.w;

  // LN2
  float mean2 = wave_sum(o0 + o1 + o2 + o3) * (1.0f / 128.0f);
  float e0 = o0 - mean2, e1 = o1 - mean2, e2 = o2 - mean2, e3 = o3 - mean2;
  float var2 = wave_sum(e0 * e0 + e1 * e1 + e2 * e2 + e3 * e3) * (1.0f / 128.0f);
  float inv2 = rsqrtf(var2 + 1e-5f);
  float4 w2v = *(const float4*)(ln2g + lane * 4);
  float4 b2v = *(const float4*)(ln2b + lane * 4);
  float4 res;
  res.x = e0 * inv2 * w2v.x + b2v.x;
  res.y = e1 * inv2 * w2v.y + b2v.y;
  res.z = e2 * inv2 * w2v.z + b2v.z;
  res.w = e3 * inv2 * w2v.w + b2v.w;
  *(float4*)(out + base) = res;
}

extern "C" void kernel_launch(void* const* d_in, const int* in_sizes, int n_in,
                              void* d_out, int out_size, void* d_ws, size_t ws_size,
                              hipStream_t stream) {
  const float*     x      = (const float*)d_in[0];
  const long long* ei     = (const long long*)d_in[1];
  const float*     degree = (const float*)d_in[2];
  const float*     W_fc   = (const float*)d_in[3];
  const float*     b_fc   = (const float*)d_in[4];
  const float*     W_rate = (const float*)d_in[5];
  const float*     W1     = (const float*)d_in[6];
  const float*     b1     = (const float*)d_in[7];
  const float*     W2     = (const float*)d_in[8];
  const float*     b2     = (const float*)d_in[9];
  const float*     ln1g   = (const float*)d_in[10];
  const float*     ln1b   = (const float*)d_in[11];
  const float*     ln2g   = (const float*)d_in[12];
  const float*     ln2b   = (const float*)d_in[13];

  const int Nn = in_sizes[0] / 128;
  const int E  = in_sizes[1] / 2;

  // Workspace layout (floats): z | rate | gamma_pre | agg | cnt(int)
  float* zbuf    = (float*)d_ws;
  float* ratebuf = zbuf    + (size_t)Nn * 128;
  float* gpre    = ratebuf + (size_t)Nn * 128;
  float* agg     = gpre    + (size_t)Nn * 128;
  int*   cnt     = (int*)(agg + (size_t)Nn * 128);

  // Zero agg + cnt (contiguous) — graph-capture-safe async memset
  hipMemsetAsync(agg, 0, ((size_t)Nn * 128 + (size_t)Nn) * sizeof(float), stream);

  dim3 b256(256);
  gemm_fused_kernel<<<(Nn + ROWS - 1) / ROWS, b256, 0, stream>>>(
      x, W_fc, b_fc, W_rate, W1, b1, W2, b2, zbuf, ratebuf, gpre, Nn);
  edge_kernel<<<(E + 7) / 8, b256, 0, stream>>>(ei, ei + E, zbuf, agg, cnt, E);
  final_kernel<<<(Nn + 7) / 8, b256, 0, stream>>>(
      zbuf, ratebuf, gpre, agg, cnt, degree, ln1g, ln1b, ln2g, ln2b,
      (float*)d_out, Nn);
}